// SetEdgeModel_36189394436993
// MI455X (gfx1250) — compile-verified
//
#include <hip/hip_runtime.h>
#include <hip/hip_bf16.h>

typedef float v2f __attribute__((ext_vector_type(2)));
typedef float v8f __attribute__((ext_vector_type(8)));

// ---------------------------------------------------------------------------
// CDNA5 fp32 WMMA: D(16x16,f32) = A(16x4,f32) x B(4x16,f32) + C
// ---------------------------------------------------------------------------
__device__ __forceinline__ v8f wmma_f32x4(v2f a, v2f b, v8f c) {
    return __builtin_amdgcn_wmma_f32_16x16x4_f32(false, a, false, b,
                                                 (short)0, c, false, false);
}

// ---------------------------------------------------------------------------
// Register-blocked GEMM: one wave computes a 64(M) x 32(N) macro-tile.
// acc[4][2] v8f = 64 accumulator VGPRs; per K-step: 4 b64 A-loads +
// 4 b32 B-loads feed 8 WMMAs (loads/WMMA = 1.0 vs 3 unblocked).
// A rows are clamped to M-1 for tail blocks (reads valid memory, results
// discarded by the store guard) so EXEC stays all-ones around every WMMA.
// grid = (ceil(M/64), N/32), block = 32 (one wave).
// ---------------------------------------------------------------------------
__global__ void gemm_wmma_f32_b(const float* __restrict__ A,
                                const float* __restrict__ B,
                                const float* __restrict__ bias,   // may be null
                                const float* __restrict__ Cin,    // may be null
                                float* __restrict__ C,
                                int M, int N, int K, int relu) {
    const int lane = threadIdx.x;          // 0..31
    const int half = lane >> 4;            // 0 or 1
    const int idx  = lane & 15;
    const int mBase = blockIdx.x * 64;
    const int nBase = blockIdx.y * 32;

    v8f acc[4][2];
    #pragma unroll
    for (int t = 0; t < 4; ++t) {
        #pragma unroll
        for (int u = 0; u < 2; ++u) {
            if (Cin) {
                #pragma unroll
                for (int v = 0; v < 8; ++v) {
                    int row = mBase + 16 * t + v + 8 * half;
                    row = row < M ? row : M - 1;
                    acc[t][u][v] = Cin[(size_t)row * N + nBase + 16 * u + idx];
                }
            } else {
                acc[t][u] = (v8f){};
            }
        }
    }

    const float* __restrict__ Arow[4];
    #pragma unroll
    for (int t = 0; t < 4; ++t) {
        int row = mBase + 16 * t + idx;
        row = row < M ? row : M - 1;
        Arow[t] = A + (size_t)row * K;
    }

    for (int k = 0; k < K; k += 4) {
        const int k0 = k + 2 * half;
        v2f a[4];
        #pragma unroll
        for (int t = 0; t < 4; ++t) {
            const float2 av = *(const float2*)(Arow[t] + k0);   // global_load_b64
            a[t].x = av.x; a[t].y = av.y;
        }
        v2f b[2];
        #pragma unroll
        for (int u = 0; u < 2; ++u) {
            b[u].x = B[(size_t)k0 * N + nBase + 16 * u + idx];
            b[u].y = B[(size_t)(k0 + 1) * N + nBase + 16 * u + idx];
        }
        #pragma unroll
        for (int t = 0; t < 4; ++t)
            #pragma unroll
            for (int u = 0; u < 2; ++u)
                acc[t][u] = wmma_f32x4(a[t], b[u], acc[t][u]);
    }

    #pragma unroll
    for (int u = 0; u < 2; ++u) {
        const float bv = bias ? bias[nBase + 16 * u + idx] : 0.0f;
        #pragma unroll
        for (int t = 0; t < 4; ++t) {
            #pragma unroll
            for (int v = 0; v < 8; ++v) {
                const int row = mBase + 16 * t + v + 8 * half;
                if (row < M) {
                    float x = acc[t][u][v] + bv;
                    if (relu) x = fmaxf(x, 0.0f);
                    C[(size_t)row * N + nBase + 16 * u + idx] = x;
                }
            }
        }
    }
}

// ---------------------------------------------------------------------------
// Edge-head GEMM1, register-blocked 64x32, with fused feature construction:
//   feat[p,c] = c<256 ? |hi[c]-hj[c]| : hi[c-256]*hj[c-256]   (K=512, N=128)
// Branchless A-construction: k0&255 indexes, VALU select picks |d| vs prod,
// so no EXEC divergence near the WMMAs. P % 64 == 0 -> no tail guards.
// grid = (P/64, 4), block = 32.
// ---------------------------------------------------------------------------
__global__ void edge_gemm_wmma_b(const float* __restrict__ Hn,
                                 const int* __restrict__ i_idx,
                                 const int* __restrict__ j_idx,
                                 const float* __restrict__ W1,
                                 const float* __restrict__ b1,
                                 float* __restrict__ Z) {
    const int lane = threadIdx.x;
    const int half = lane >> 4;
    const int idx  = lane & 15;
    const int mBase = blockIdx.x * 64;
    const int nBase = blockIdx.y * 32;
    const int NN = 128, KK = 512, H = 256;

    const float* __restrict__ hi[4];
    const float* __restrict__ hj[4];
    #pragma unroll
    for (int t = 0; t < 4; ++t) {
        const int p = mBase + 16 * t + idx;
        hi[t] = Hn + (size_t)i_idx[p] * H;
        hj[t] = Hn + (size_t)j_idx[p] * H;
    }

    v8f acc[4][2];
    #pragma unroll
    for (int t = 0; t < 4; ++t)
        #pragma unroll
        for (int u = 0; u < 2; ++u)
            acc[t][u] = (v8f){};

    for (int k = 0; k < KK; k += 4) {
        const int k0 = k + 2 * half;        // even; pair never straddles 256
        const int c  = k0 & 255;
        const bool absPart = (k0 < 256);
        v2f a[4];
        #pragma unroll
        for (int t = 0; t < 4; ++t) {
            const float2 x = *(const float2*)(hi[t] + c);
            const float2 y = *(const float2*)(hj[t] + c);
            a[t].x = absPart ? fabsf(x.x - y.x) : x.x * y.x;
            a[t].y = absPart ? fabsf(x.y - y.y) : x.y * y.y;
        }
        v2f b[2];
        #pragma unroll
        for (int u = 0; u < 2; ++u) {
            b[u].x = W1[(size_t)k0 * NN + nBase + 16 * u + idx];
            b[u].y = W1[(size_t)(k0 + 1) * NN + nBase + 16 * u + idx];
        }
        #pragma unroll
        for (int t = 0; t < 4; ++t)
            #pragma unroll
            for (int u = 0; u < 2; ++u)
                acc[t][u] = wmma_f32x4(a[t], b[u], acc[t][u]);
    }

    #pragma unroll
    for (int u = 0; u < 2; ++u) {
        const float bv = b1[nBase + 16 * u + idx];
        #pragma unroll
        for (int t = 0; t < 4; ++t) {
            #pragma unroll
            for (int v = 0; v < 8; ++v) {
                const int row = mBase + 16 * t + v + 8 * half;
                Z[(size_t)row * NN + nBase + 16 * u + idx] =
                    fmaxf(acc[t][u][v] + bv, 0.0f);
            }
        }
    }
}

// ---------------------------------------------------------------------------
// Elementwise / graph kernels
// ---------------------------------------------------------------------------
__global__ void zero_f32(float* __restrict__ p, long long n) {
    long long i = (long long)blockIdx.x * blockDim.x + threadIdx.x;
    long long stride = (long long)gridDim.x * blockDim.x;
    for (; i < n; i += stride) p[i] = 0.0f;
}

__global__ void degree_kernel(const int* __restrict__ dst, float* __restrict__ deg, int E) {
    int e = blockIdx.x * blockDim.x + threadIdx.x;
    if (e < E) atomicAdd(&deg[dst[e]], 1.0f);
}

// One block (256 threads) per edge: agg[dst[e], :] += h[src[e], :]
__global__ void scatter_add_kernel(const float* __restrict__ h,
                                   const int* __restrict__ src,
                                   const int* __restrict__ dst,
                                   float* __restrict__ agg) {
    const int e = blockIdx.x;
    const int f = threadIdx.x;          // 0..255
    const int s = src[e], d = dst[e];
    atomicAdd(&agg[(size_t)d * 256 + f], h[(size_t)s * 256 + f]);
}

__global__ void mean_div_kernel(float* __restrict__ agg,
                                const float* __restrict__ deg, long long n) {
    long long i = (long long)blockIdx.x * blockDim.x + threadIdx.x;
    if (i < n) agg[i] /= fmaxf(deg[i >> 8], 1.0f);
}

// h[row,:] = LayerNorm(relu(in[row,:])) * gamma + beta  (row width 256)
__global__ void relu_ln_kernel(const float* __restrict__ in,
                               const float* __restrict__ gamma,
                               const float* __restrict__ beta,
                               float* __restrict__ out) {
    __shared__ float red[256];
    const int row = blockIdx.x;
    const int f = threadIdx.x;
    float x = fmaxf(in[(size_t)row * 256 + f], 0.0f);

    red[f] = x; __syncthreads();
    for (int s = 128; s > 0; s >>= 1) {
        if (f < s) red[f] += red[f + s];
        __syncthreads();
    }
    const float mu = red[0] * (1.0f / 256.0f);
    __syncthreads();

    const float d = x - mu;
    red[f] = d * d; __syncthreads();
    for (int s = 128; s > 0; s >>= 1) {
        if (f < s) red[f] += red[f + s];
        __syncthreads();
    }
    const float var = red[0] * (1.0f / 256.0f);
    out[(size_t)row * 256 + f] = d * rsqrtf(var + 1e-5f) * gamma[f] + beta[f];
}

// H[row,:] = h[row,:] / max(||h[row,:]||2, 1e-12)
__global__ void l2norm_kernel(const float* __restrict__ h, float* __restrict__ H) {
    __shared__ float red[256];
    const int row = blockIdx.x;
    const int f = threadIdx.x;
    const float x = h[(size_t)row * 256 + f];
    red[f] = x * x; __syncthreads();
    for (int s = 128; s > 0; s >>= 1) {
        if (f < s) red[f] += red[f + s];
        __syncthreads();
    }
    const float inv = 1.0f / fmaxf(sqrtf(red[0]), 1e-12f);
    H[(size_t)row * 256 + f] = x * inv;
}

// logits[p] = z2[p,:] . W3 + b3   (wave32 shuffle reduction, 8 rows / block)
__global__ void logits_kernel(const float* __restrict__ z2,
                              const float* __restrict__ W3,
                              const float* __restrict__ b3,
                              float* __restrict__ out, int P) {
    const int wavesPerBlock = blockDim.x >> 5;
    const int p = blockIdx.x * wavesPerBlock + (threadIdx.x >> 5);
    const int lane = threadIdx.x & 31;
    if (p >= P) return;
    float s = 0.0f;
    #pragma unroll
    for (int k = lane; k < 128; k += 32) s += z2[(size_t)p * 128 + k] * W3[k];
    #pragma unroll
    for (int off = 16; off > 0; off >>= 1) s += __shfl_xor(s, off, 32);
    if (lane == 0) out[p] = s + b3[0];
}

// ---------------------------------------------------------------------------
// Host-side orchestration
// ---------------------------------------------------------------------------
extern "C" void kernel_launch(void* const* d_in, const int* in_sizes, int n_in,
                              void* d_out, int out_size, void* d_ws, size_t ws_size,
                              hipStream_t stream) {
    const int IN_DIM = 480, HD = 256, HID = 128;
    const int N = in_sizes[0] / IN_DIM;       // 50000
    const int E = in_sizes[1] / 2;            // 300000
    const int P = in_sizes[2];                // 200000

    const float* X        = (const float*)d_in[0];
    const int*   edge     = (const int*)  d_in[1];
    const int*   i_idx    = (const int*)  d_in[2];
    const int*   j_idx    = (const int*)  d_in[3];
    const float* W_in     = (const float*)d_in[4];
    const float* b_in     = (const float*)d_in[5];
    const float* Ws_self  = (const float*)d_in[6];   // [3,256,256]
    const float* bs_self  = (const float*)d_in[7];   // [3,256]
    const float* Ws_nei   = (const float*)d_in[8];
    const float* bs_nei   = (const float*)d_in[9];
    const float* gammas   = (const float*)d_in[10];
    const float* betas    = (const float*)d_in[11];
    const float* W1       = (const float*)d_in[12];  // [512,128]
    const float* b1       = (const float*)d_in[13];
    const float* W2       = (const float*)d_in[14];  // [128,128]
    const float* b2       = (const float*)d_in[15];
    const float* W3       = (const float*)d_in[16];  // [128]
    const float* b3       = (const float*)d_in[17];

    const int* src = edge;          // edge_index[0]
    const int* dst = edge + E;      // edge_index[1]

    // workspace layout (floats): h | agg | tmp | z2 | deg
    float* ws   = (float*)d_ws;
    const size_t NH = (size_t)N * HD;             // 12.8M
    float* h    = ws;                             // [N,256]
    float* agg  = h   + NH;                       // [N,256]
    float* tmp  = agg + NH;                       // [N,256]  (contig after agg)
    float* z2   = tmp + NH;                       // [P,128]
    float* deg  = z2  + (size_t)P * HID;          // [N]
    float* z1   = agg;                            // reuse agg+tmp post-loop

    float* Hout   = (float*)d_out;                // [N,256]
    float* logits = Hout + NH;                    // [P]

    const dim3 wmmaBlk(32, 1, 1);
    const int mBlocksN = (N + 63) / 64;           // 782 (tail-clamped)
    const int mBlocksP = P / 64;                  // 3125 (exact)

    // degree (constant across layers)
    zero_f32<<<1024, 256, 0, stream>>>(deg, (long long)N);
    degree_kernel<<<(E + 255) / 256, 256, 0, stream>>>(dst, deg, E);

    // encoder: h = relu(X @ W_in + b_in)
    gemm_wmma_f32_b<<<dim3(mBlocksN, HD / 32), wmmaBlk, 0, stream>>>(
        X, W_in, b_in, nullptr, h, N, HD, IN_DIM, 1);

    for (int l = 0; l < 3; ++l) {
        zero_f32<<<4096, 256, 0, stream>>>(agg, (long long)NH);
        scatter_add_kernel<<<E, 256, 0, stream>>>(h, src, dst, agg);
        mean_div_kernel<<<(int)((NH + 255) / 256), 256, 0, stream>>>(agg, deg, (long long)NH);

        // tmp = h @ Ws_self[l] + bs_self[l]
        gemm_wmma_f32_b<<<dim3(mBlocksN, HD / 32), wmmaBlk, 0, stream>>>(
            h, Ws_self + (size_t)l * HD * HD, bs_self + (size_t)l * HD,
            nullptr, tmp, N, HD, HD, 0);
        // tmp += mean @ Ws_nei[l] + bs_nei[l]
        gemm_wmma_f32_b<<<dim3(mBlocksN, HD / 32), wmmaBlk, 0, stream>>>(
            agg, Ws_nei + (size_t)l * HD * HD, bs_nei + (size_t)l * HD,
            tmp, tmp, N, HD, HD, 0);
        // h = LN(relu(tmp))
        relu_ln_kernel<<<N, 256, 0, stream>>>(
            tmp, gammas + (size_t)l * HD, betas + (size_t)l * HD, h);
    }

    // Hout = L2-normalize(h)  -> first N*256 floats of d_out
    l2norm_kernel<<<N, 256, 0, stream>>>(h, Hout);

    // z1 = relu(feat @ W1 + b1), feat built on the fly from Hout
    edge_gemm_wmma_b<<<dim3(mBlocksP, HID / 32), wmmaBlk, 0, stream>>>(
        Hout, i_idx, j_idx, W1, b1, z1);

    // z2 = relu(z1 @ W2 + b2)
    gemm_wmma_f32_b<<<dim3(mBlocksP, HID / 32), wmmaBlk, 0, stream>>>(
        z1, W2, b2, nullptr, z2, P, HID, HID, 1);

    // logits
    logits_kernel<<<(P + 7) / 8, 256, 0, stream>>>(z2, W3, b3, logits, P);

    (void)n_in; (void)out_size; (void)ws_size;
}